// RouteGraphEncoder_71811853189760
// MI455X (gfx1250) — compile-verified
//
#include <hip/hip_runtime.h>
#include <hip/hip_bf16.h>

#define LL  2048
#define BB  64
#define DD  256
#define HH  4
#define HDD 64
#define EE  4096

typedef __attribute__((ext_vector_type(2))) float    v2f;
typedef __attribute__((ext_vector_type(8))) float    v8f;
typedef __attribute__((ext_vector_type(4))) unsigned v4u;
typedef __attribute__((ext_vector_type(8))) int      v8i;
typedef __attribute__((ext_vector_type(4))) int      v4i;

// ---------------------------------------------------------------------------
// hh[b] = h[b] @ W  via V_WMMA_F32_16X16X4_F32.
// W (256x256 f32, 256KB) is DMA'd into LDS once per workgroup by the Tensor
// Data Mover, then B-fragments come from LDS. Each block (8 waves) processes
// 4 consecutive 16-row M-tiles; each wave owns a 32-column strip (2 N-tiles).
// grid.x = B*(L/16)/4 = 2048, block = 256, dynamic LDS = 256KB.
// ---------------------------------------------------------------------------
__global__ void gemm_wmma_f32(const float* __restrict__ h,
                              const float* __restrict__ W,
                              float* __restrict__ hh) {
    extern __shared__ float Wlds[];   // DD*DD floats at LDS offset 0

    const int lane = threadIdx.x & 31;
    const int wave = threadIdx.x >> 5;
    const int half = lane >> 4;       // 0: lanes 0-15, 1: lanes 16-31
    const int l16  = lane & 15;

    // ---- TDM: DMA W -> LDS (wave 0 issues; EXEC ignored, wave-level op) ----
    if (threadIdx.x < 32) {
        const unsigned long long gaddr = (unsigned long long)(uintptr_t)W;
        v4u g0;
        g0[0] = 1u;                                   // count=1 (valid user D#)
        g0[1] = 0u;                                   // lds_addr = 0
        g0[2] = (unsigned)(gaddr & 0xFFFFFFFFu);      // global_addr[31:0]
        g0[3] = (unsigned)((gaddr >> 32) & 0x01FFFFFFu) | (2u << 30); // addr[56:32], type=2

        v8i g1;
        g1[0] = (int)(2u << 16);                      // data_size = 4B
        g1[1] = (int)(256u << 16);                    // tensor_dim0[15:0] = 256
        g1[2] = (int)(256u << 16);                    // dim0[31:16]=0, tensor_dim1[15:0]=256
        g1[3] = (int)(256u << 16);                    // dim1[31:16]=0, tile_dim0=256
        g1[4] = (int)256;                             // tile_dim1=256, tile_dim2=0
        g1[5] = (int)256;                             // tensor_dim0_stride[31:0]=256
        g1[6] = 0;                                    // stride0 hi, dim1_stride lo
        g1[7] = 0;                                    // dim1_stride hi (unused, 2-D)

        v4i gz4 = {0, 0, 0, 0};                       // groups 2/3 unused (2-D)
        v8i gz8 = {0, 0, 0, 0, 0, 0, 0, 0};
        __builtin_amdgcn_tensor_load_to_lds(g0, g1, gz4, gz4, gz8, 0);
        __builtin_amdgcn_s_wait_tensorcnt(0);
    }
    __syncthreads();

    const int n0 = wave * 32;                          // 32-col strip per wave
#pragma unroll 1
    for (int t = 0; t < 4; ++t) {
        const int gt = blockIdx.x * 4 + t;             // global M-tile id
        const int b  = gt >> 7;                        // / (L/16)
        const int mt = gt & 127;
        const float* A = h  + ((size_t)(b * LL + mt * 16)) * DD;
        float*       O = hh + ((size_t)(b * LL + mt * 16)) * DD;

        v8f c0 = {}, c1 = {};
        for (int k0 = 0; k0 < DD; k0 += 4) {
            const int k = k0 + 2 * half;
            v2f av, b0, b1;
            // A 16x4 f32: lane half selects K pair {k0,k0+1} / {k0+2,k0+3}
            av.x = A[l16 * DD + k];
            av.y = A[l16 * DD + k + 1];
            // B 4x16 f32 from LDS: VGPR v, half h -> row K=2h+v, col n
            b0.x = Wlds[k * DD + n0 + l16];
            b0.y = Wlds[(k + 1) * DD + n0 + l16];
            b1.x = Wlds[k * DD + n0 + 16 + l16];
            b1.y = Wlds[(k + 1) * DD + n0 + 16 + l16];
            c0 = __builtin_amdgcn_wmma_f32_16x16x4_f32(false, av, false, b0,
                                                       (short)0, c0, false, false);
            c1 = __builtin_amdgcn_wmma_f32_16x16x4_f32(false, av, false, b1,
                                                       (short)0, c1, false, false);
        }
        // D 16x16: VGPR r, half -> row r + 8*half, col
#pragma unroll
        for (int r = 0; r < 8; ++r) {
            O[(size_t)(r + 8 * half) * DD + n0 + l16]      = c0[r];
            O[(size_t)(r + 8 * half) * DD + n0 + 16 + l16] = c1[r];
        }
    }
}

// ---------------------------------------------------------------------------
// One wave per (batch, edge): attention scores -> exp -> denom segment-sum.
// ---------------------------------------------------------------------------
__global__ void gat_score(const float* __restrict__ hh,
                          const int*   __restrict__ edge_index,
                          const float* __restrict__ a,
                          float* __restrict__ exbuf,
                          float* __restrict__ denom) {
    const int gw   = blockIdx.x * 8 + (threadIdx.x >> 5);
    const int lane = threadIdx.x & 31;
    const int b    = gw / EE;
    const int e    = gw % EE;
    const int src  = edge_index[(size_t)b * 2 * EE + e];
    const int dst  = edge_index[(size_t)b * 2 * EE + EE + e];
    const int head = lane >> 3;
    const int j0   = (lane & 7) * 8;

    const float* hs = hh + ((size_t)b * LL + src) * DD + head * HDD + j0;
    const float* hd = hh + ((size_t)b * LL + dst) * DD + head * HDD + j0;
    const float* as = a + (size_t)head * 2 * HDD + j0;
    const float* ad = as + HDD;

    float p = 0.f;
#pragma unroll
    for (int j = 0; j < 8; ++j)
        p += hs[j] * as[j] + hd[j] * ad[j];
    p += __shfl_xor(p, 1, 32);
    p += __shfl_xor(p, 2, 32);
    p += __shfl_xor(p, 4, 32);

    if ((lane & 7) == 0) {
        float s  = (p > 0.f) ? p : 0.2f * p;   // leaky_relu(0.2)
        float ex = expf(s);
        exbuf[((size_t)b * EE + e) * HH + head] = ex;
        atomicAdd(&denom[((size_t)b * LL + dst) * HH + head], ex);
    }
}

// ---------------------------------------------------------------------------
// One wave per (batch, edge): acc[dst] += hh[src] * attn  (8 floats per lane).
// ---------------------------------------------------------------------------
__global__ void gat_aggregate(const float* __restrict__ hh,
                              const int*   __restrict__ edge_index,
                              const float* __restrict__ exbuf,
                              const float* __restrict__ denom,
                              float* __restrict__ acc) {
    const int gw   = blockIdx.x * 8 + (threadIdx.x >> 5);
    const int lane = threadIdx.x & 31;
    const int b    = gw / EE;
    const int e    = gw % EE;
    const int src  = edge_index[(size_t)b * 2 * EE + e];
    const int dst  = edge_index[(size_t)b * 2 * EE + EE + e];
    const int head = lane >> 3;
    const int d0   = lane * 8;

    const float attn = exbuf[((size_t)b * EE + e) * HH + head] /
                       (denom[((size_t)b * LL + dst) * HH + head] + 1e-16f);

    const float* hs = hh  + ((size_t)b * LL + src) * DD + d0;
    float*       ac = acc + ((size_t)b * LL + dst) * DD + d0;
#pragma unroll
    for (int j = 0; j < 8; ++j)
        atomicAdd(&ac[j], hs[j] * attn);
}

// ---------------------------------------------------------------------------
// One wave per row: h = relu(layernorm(h + acc) * g + beta), in place.
// ---------------------------------------------------------------------------
__global__ void ln_relu(const float* __restrict__ acc,
                        const float* __restrict__ g,
                        const float* __restrict__ beta,
                        float* __restrict__ h) {
    const int row  = blockIdx.x * 8 + (threadIdx.x >> 5);
    const int lane = threadIdx.x & 31;
    const int d0   = lane * 8;

    float*       xr = h   + (size_t)row * DD + d0;
    const float* ar = acc + (size_t)row * DD + d0;

    float v[8];
    float s = 0.f;
#pragma unroll
    for (int j = 0; j < 8; ++j) { v[j] = xr[j] + ar[j]; s += v[j]; }
#pragma unroll
    for (int m = 1; m < 32; m <<= 1) s += __shfl_xor(s, m, 32);
    const float mu = s * (1.0f / DD);

    float s2 = 0.f;
#pragma unroll
    for (int j = 0; j < 8; ++j) { float t = v[j] - mu; s2 += t * t; }
#pragma unroll
    for (int m = 1; m < 32; m <<= 1) s2 += __shfl_xor(s2, m, 32);
    const float var = s2 * (1.0f / DD);
    const float rs  = rsqrtf(var + 1e-5f);

#pragma unroll
    for (int j = 0; j < 8; ++j) {
        float y = (v[j] - mu) * rs * g[d0 + j] + beta[d0 + j];
        xr[j] = (y > 0.f) ? y : 0.f;
    }
}

// ---------------------------------------------------------------------------
// Layout transposes: (L,B,D) <-> (B,L,D), float4-vectorized.
// ---------------------------------------------------------------------------
__global__ void transpose_in(const float4* __restrict__ in, float4* __restrict__ h) {
    const int DQ = DD / 4;
    size_t tid = (size_t)blockIdx.x * blockDim.x + threadIdx.x;
    int    d   = (int)(tid % DQ);
    size_t row = tid / DQ;                 // row = l*B + b  (coalesced reads)
    int    b   = (int)(row % BB);
    int    l   = (int)(row / BB);
    h[((size_t)b * LL + l) * DQ + d] = in[row * DQ + d];
}

__global__ void transpose_out(const float4* __restrict__ h, float4* __restrict__ out) {
    const int DQ = DD / 4;
    size_t tid = (size_t)blockIdx.x * blockDim.x + threadIdx.x;
    int    d   = (int)(tid % DQ);
    size_t row = tid / DQ;                 // row = b*L + l  (coalesced reads)
    int    l   = (int)(row % LL);
    int    b   = (int)(row / LL);
    out[((size_t)l * BB + b) * DQ + d] = h[row * DQ + d];
}

// ---------------------------------------------------------------------------
// Host launcher. Inputs: 0 route_emb, 1 route_len (unused), 2 edge_index,
// 3 W0, 4 a0, 5 g0, 6 b0, 7 W1, 8 a1, 9 g1, 10 b1. Output f32 (L,B,D).
// ---------------------------------------------------------------------------
extern "C" void kernel_launch(void* const* d_in, const int* in_sizes, int n_in,
                              void* d_out, int out_size, void* d_ws, size_t ws_size,
                              hipStream_t stream) {
    (void)in_sizes; (void)n_in; (void)out_size; (void)ws_size;

    const float* route_emb = (const float*)d_in[0];
    const int*   edge_idx  = (const int*)d_in[2];
    const float* Wl[2]  = {(const float*)d_in[3], (const float*)d_in[7]};
    const float* al[2]  = {(const float*)d_in[4], (const float*)d_in[8]};
    const float* gl[2]  = {(const float*)d_in[5], (const float*)d_in[9]};
    const float* bl[2]  = {(const float*)d_in[6], (const float*)d_in[10]};

    const size_t NBLD = (size_t)BB * LL * DD;
    float* hbuf  = (float*)d_ws;                   // (B,L,D) current features
    float* hhbuf = hbuf + NBLD;                    // (B,L,D) h @ W
    float* exbuf = hhbuf + NBLD;                   // (B,E,H) exp(scores)
    float* denom = exbuf + (size_t)BB * EE * HH;   // (B,L,H) segment sums
    float* acc   = (float*)d_out;                  // (B,L,D) scratch; rewritten at end

    transpose_in<<<(unsigned)(NBLD / 4 / 256), 256, 0, stream>>>(
        (const float4*)route_emb, (float4*)hbuf);

    const unsigned gemm_grid = BB * (LL / 16) / 4;          // 2048 blocks
    const size_t   gemm_lds  = (size_t)DD * DD * sizeof(float);  // 256 KB

    for (int layer = 0; layer < 2; ++layer) {
        gemm_wmma_f32<<<gemm_grid, 256, gemm_lds, stream>>>(hbuf, Wl[layer], hhbuf);

        (void)hipMemsetAsync(denom, 0, (size_t)BB * LL * HH * sizeof(float), stream);
        (void)hipMemsetAsync(acc,   0, NBLD * sizeof(float), stream);

        gat_score<<<BB * EE / 8, 256, 0, stream>>>(hhbuf, edge_idx, al[layer],
                                                   exbuf, denom);
        gat_aggregate<<<BB * EE / 8, 256, 0, stream>>>(hhbuf, edge_idx, exbuf,
                                                       denom, acc);
        ln_relu<<<BB * LL / 8, 256, 0, stream>>>(acc, gl[layer], bl[layer], hbuf);
    }

    transpose_out<<<(unsigned)(NBLD / 4 / 256), 256, 0, stream>>>(
        (const float4*)hbuf, (float4*)d_out);
}